// Net_87454124081304
// MI455X (gfx1250) — compile-verified
//
#include <hip/hip_runtime.h>
#include <hip/hip_bf16.h>

#define GN 500000
#define GE 8000000
#define PK1 5000
#define PK2 500
#define PK3 50

typedef float v2f __attribute__((ext_vector_type(2)));
typedef float v8f __attribute__((ext_vector_type(8)));

// ---------------- utility kernels ----------------
__global__ void zero_f32(float* p, int n) {
  int i = blockIdx.x * blockDim.x + threadIdx.x;
  if (i < n) p[i] = 0.f;
}
__global__ void zero_u32(unsigned* p, int n) {
  int i = blockIdx.x * blockDim.x + threadIdx.x;
  if (i < n) p[i] = 0u;
}

__device__ __forceinline__ void atomAddF(float* p, float v) {
  unsafeAtomicAdd(p, v);   // global_atomic_add_f32 on gfx1250
}

// ---------------- operand packing ----------------
__global__ void pack_x(const float* __restrict__ X, float* __restrict__ Xp,
                       int n, int fin, int lda) {
  int i = blockIdx.x * blockDim.x + threadIdx.x;
  if (i >= n * lda) return;
  int row = i / lda, k = i - row * lda;
  Xp[i] = (k < fin) ? X[(size_t)row * fin + k] : 0.f;
}
__global__ void pack_wt(const float* __restrict__ W, float* __restrict__ Wt,
                        int fin, int fout, int ldb, int fout_pad) {
  int i = blockIdx.x * blockDim.x + threadIdx.x;
  if (i >= fout_pad * ldb) return;
  int c = i / ldb, k = i - c * ldb;
  Wt[i] = (c < fout && k < fin) ? W[(size_t)k * fout + c] : 0.f;
}

// ---------------- WMMA GEMM: fully branch-free ----------------
// Xp: [round16(n), LDK] zero-padded.  Wt: [round16(FOUT), LDK] packed col-major.
// Out: [round16(n), LDO] padded output (LDO = round16(FOUT)) -> unconditional stores.
template <int LDK, int FOUT, int LDO>
__global__ void gemm_wmma(const float* __restrict__ Xp, const float* __restrict__ Wt,
                          float* __restrict__ Out, int n) {
  int wave = threadIdx.x >> 5;
  int lane = threadIdx.x & 31;
  int r0 = (blockIdx.x * 4 + wave) * 16;
  if (r0 >= ((n + 15) & ~15)) return;       // wave-uniform: EXEC stays all-ones
  int c0 = blockIdx.y * 16;
  int m = lane & 15, half = lane >> 4;

  const v2f* A = (const v2f*)(Xp + (size_t)(r0 + m) * LDK);
  const v2f* B = (const v2f*)(Wt + (size_t)(c0 + m) * LDK);
  v8f acc = {};
#pragma unroll
  for (int ks = 0; ks < LDK / 4; ++ks) {
    v2f a = A[2 * ks + half];               // global_load_b64
    v2f b = B[2 * ks + half];
    acc = __builtin_amdgcn_wmma_f32_16x16x4_f32(false, a, false, b,
                                                (short)0, acc, false, false);
  }
  float* O = Out + (size_t)(r0 + 8 * half) * LDO + (c0 + m);
#pragma unroll
  for (int v = 0; v < 8; ++v) O[(size_t)v * LDO] = acc[v];   // unconditional
}

// ---------------- graph kernels ----------------
__global__ void init_edges(const int* __restrict__ ei, int* s, int* d, float* w, int E) {
  int e = blockIdx.x * blockDim.x + threadIdx.x;
  if (e >= E) return;
  s[e] = ei[e];
  d[e] = ei[E + e];
  w[e] = 1.f;
}

__global__ void deg_add(const int* __restrict__ dst, const float* __restrict__ ew,
                        float* deg, int E) {
  int e = blockIdx.x * blockDim.x + threadIdx.x;
  if (e >= E) return;
  float w = ew[e];
  if (w != 0.f) atomAddF(&deg[dst[e]], w);
}

template <int FOUT, int LDO>
__global__ void edge_scatter(const int* __restrict__ src, const int* __restrict__ dst,
                             const float* __restrict__ ew, const float* __restrict__ deg,
                             const float* __restrict__ hW, float* agg, int E) {
  int e = blockIdx.x * blockDim.x + threadIdx.x;
  if (e >= E) return;
  float w = ew[e];
  if (w == 0.f) return;
  int s = src[e], d = dst[e];
  float nrm = w * rsqrtf(deg[s] + 1.f) * rsqrtf(deg[d] + 1.f);
  const float* hs = hW + (size_t)s * LDO;
  float* ad = agg + (size_t)d * LDO;
#pragma unroll
  for (int f = 0; f < FOUT; ++f) atomAddF(&ad[f], hs[f] * nrm);
}

template <int FOUT, int LDO>
__global__ void gcn_finalize(float* hW, const float* __restrict__ agg,
                             const float* __restrict__ deg, const float* __restrict__ bias,
                             int n, int relu) {
  int i = blockIdx.x * blockDim.x + threadIdx.x;
  if (i >= n * FOUT) return;
  int node = i / FOUT, f = i - node * FOUT;
  size_t idx = (size_t)node * LDO + f;
  float di = 1.f / (deg[node] + 1.f);       // dinv*dinv = 1/deg
  float v = agg[idx] + hW[idx] * di + bias[f];
  if (relu) v = fmaxf(v, 0.f);
  hW[idx] = v;
}

// ---------------- top-k pooling ----------------
__global__ void score_kernel(const float* __restrict__ X, int stride,
                             const float* __restrict__ p, int f, int n, float* score) {
  int i = blockIdx.x * blockDim.x + threadIdx.x;
  if (i >= n) return;
  float nrm = 0.f;
  for (int j = 0; j < f; ++j) nrm += p[j] * p[j];
  nrm = rsqrtf(nrm);
  float s = 0.f;
  const float* xi = X + (size_t)i * stride;
  for (int j = 0; j < f; ++j) s += xi[j] * p[j];
  score[i] = s * nrm;
}

// state words: 0=lo(f32) 1=hi(f32) 2=cnt 3=ticket 4=minenc 5=maxenc
__device__ __forceinline__ unsigned encF(float f) {
  unsigned u = __float_as_uint(f);
  return (u & 0x80000000u) ? ~u : (u | 0x80000000u);
}
__device__ __forceinline__ float decF(unsigned u) {
  unsigned x = (u & 0x80000000u) ? (u & 0x7FFFFFFFu) : ~u;
  return __uint_as_float(x);
}

__global__ void minmax_init(unsigned* st) { st[4] = 0xFFFFFFFFu; st[5] = 0u; }
__global__ void minmax_reduce(const float* __restrict__ score, int n, unsigned* st) {
  int i = blockIdx.x * blockDim.x + threadIdx.x;
  if (i >= n) return;
  unsigned e = encF(score[i]);
  atomicMin(&st[4], e);
  atomicMax(&st[5], e);
}
__global__ void bisect_init(unsigned* st) {
  st[0] = __float_as_uint(decF(st[4]) - 1.f);
  st[1] = __float_as_uint(decF(st[5]));
  st[3] = 0u;
}
__global__ void bisect_pre(unsigned* st) { st[2] = 0u; }
__global__ void bisect_count(const float* __restrict__ score, int n, unsigned* st) {
  int i = blockIdx.x * blockDim.x + threadIdx.x;
  if (i >= n) return;
  float mid = 0.5f * (__uint_as_float(st[0]) + __uint_as_float(st[1]));
  if (score[i] > mid) atomicAdd(&st[2], 1u);
}
__global__ void bisect_post(unsigned* st, int k) {
  float lo = __uint_as_float(st[0]), hi = __uint_as_float(st[1]);
  float mid = 0.5f * (lo + hi);
  if (st[2] > (unsigned)k) st[0] = __float_as_uint(mid);
  else                     st[1] = __float_as_uint(mid);
}
__global__ void select_gt(const float* __restrict__ score, int n, unsigned* st,
                          int* perm, int k) {
  int i = blockIdx.x * blockDim.x + threadIdx.x;
  if (i >= n) return;
  if (score[i] > __uint_as_float(st[1])) {
    unsigned pos = atomicAdd(&st[3], 1u);
    if (pos < (unsigned)k) perm[pos] = i;
  }
}
__global__ void select_rest(const float* __restrict__ score, int n, unsigned* st,
                            int* perm, int k) {
  int i = blockIdx.x * blockDim.x + threadIdx.x;
  if (i >= n) return;
  float s = score[i];
  if (s <= __uint_as_float(st[1]) && s > __uint_as_float(st[0])) {
    unsigned pos = atomicAdd(&st[3], 1u);
    if (pos < (unsigned)k) perm[pos] = i;
  }
}

__global__ void pool_apply(const float* __restrict__ X, int in_stride,
                           const float* __restrict__ score, const int* __restrict__ perm,
                           float* Xk, int out_stride,
                           int* keep, int* nidx, int k, int f) {
  int j = blockIdx.x * blockDim.x + threadIdx.x;
  if (j >= k) return;
  int i = perm[j];
  float t = tanhf(score[i]);
  const float* xi = X + (size_t)i * in_stride;
  float* xo = Xk + (size_t)j * out_stride;
  for (int q = 0; q < f; ++q) xo[q] = xi[q] * t;
  keep[i] = 1;
  nidx[i] = j;
}

__global__ void reindex_edges(int* s, int* d, float* w,
                              const int* __restrict__ keep, const int* __restrict__ nidx,
                              int E) {
  int e = blockIdx.x * blockDim.x + threadIdx.x;
  if (e >= E) return;
  int a = s[e], b = d[e];
  int valid = keep[a] & keep[b];
  s[e] = valid ? nidx[a] : 0;
  d[e] = valid ? nidx[b] : 0;
  w[e] = valid ? w[e] : 0.f;
}

// ---------------- final linear + log_softmax ----------------
__global__ void final_head(const float* __restrict__ H, int stride,
                           const float* __restrict__ Wl, const float* __restrict__ bl,
                           float* out, int n, int fin) {
  int r = blockIdx.x * blockDim.x + threadIdx.x;
  if (r >= n) return;
  float v[8];
#pragma unroll
  for (int c = 0; c < 8; ++c) {
    float s = bl[c];
    for (int f = 0; f < fin; ++f) s += H[(size_t)r * stride + f] * Wl[(size_t)f * 8 + c];
    v[c] = fmaxf(s, 0.f);
  }
  float mx = v[0];
#pragma unroll
  for (int c = 1; c < 8; ++c) mx = fmaxf(mx, v[c]);
  float sum = 0.f;
#pragma unroll
  for (int c = 0; c < 8; ++c) sum += expf(v[c] - mx);
  float ls = logf(sum);
#pragma unroll
  for (int c = 0; c < 8; ++c) out[(size_t)r * 8 + c] = v[c] - mx - ls;
}

// ---------------- host helpers ----------------
static inline dim3 g(long long n) { return dim3((unsigned)((n + 255) / 256)); }

template <int LDK, int FOUT, int LDO>
static void run_conv(const float* Xp, const float* Wt, const float* bias, int n, int relu,
                     float* hW, float* agg,
                     const int* ESRC, const int* EDST, const float* EEW, float* DEG,
                     hipStream_t stream) {
  int rowTiles = (n + 15) / 16;
  dim3 grid((rowTiles + 3) / 4, (FOUT + 15) / 16);
  gemm_wmma<LDK, FOUT, LDO><<<grid, dim3(128), 0, stream>>>(Xp, Wt, hW, n);
  zero_f32<<<g(n), dim3(256), 0, stream>>>(DEG, n);
  deg_add<<<g(GE), dim3(256), 0, stream>>>(EDST, EEW, DEG, GE);
  zero_f32<<<g((long long)n * LDO), dim3(256), 0, stream>>>(agg, n * LDO);
  edge_scatter<FOUT, LDO><<<g(GE), dim3(256), 0, stream>>>(ESRC, EDST, EEW, DEG, hW, agg, GE);
  gcn_finalize<FOUT, LDO><<<g((long long)n * FOUT), dim3(256), 0, stream>>>(hW, agg, DEG, bias, n, relu);
}

// ---------------- entry ----------------
extern "C" void kernel_launch(void* const* d_in, const int* in_sizes, int n_in,
                              void* d_out, int out_size, void* d_ws, size_t ws_size,
                              hipStream_t stream) {
  const float* x  = (const float*)d_in[0];
  const int*   ei = (const int*)d_in[1];
  const float* W1 = (const float*)d_in[2];  const float* b1 = (const float*)d_in[3];
  const float* p1 = (const float*)d_in[4];
  const float* W2 = (const float*)d_in[5];  const float* b2 = (const float*)d_in[6];
  const float* p2 = (const float*)d_in[7];
  const float* W3 = (const float*)d_in[8];  const float* b3 = (const float*)d_in[9];
  const float* p3 = (const float*)d_in[10];
  const float* W4 = (const float*)d_in[11]; const float* b4 = (const float*)d_in[12];
  const float* Wl = (const float*)d_in[13]; const float* bl = (const float*)d_in[14];
  float* out = (float*)d_out;

  char* ws = (char*)d_ws;
  size_t o = 0;
  auto take = [&](size_t bytes) { size_t c = o; o = (o + bytes + 255) & ~(size_t)255; return c; };
  int*      ESRC = (int*)     (ws + take((size_t)GE * 4));
  int*      EDST = (int*)     (ws + take((size_t)GE * 4));
  float*    EEW  = (float*)   (ws + take((size_t)GE * 4));
  float*    XP   = (float*)   (ws + take((size_t)GN * 8 * 4));    // packed x, lda=8
  float*    HA   = (float*)   (ws + take((size_t)GN * 32 * 4));   // conv1 hW, stride 32
  float*    HB   = (float*)   (ws + take((size_t)GN * 32 * 4));   // conv1 agg, stride 32
  float*    DEG  = (float*)   (ws + take((size_t)GN * 4));
  float*    SCORE= (float*)   (ws + take((size_t)GN * 4));
  int*      KEEP = (int*)     (ws + take((size_t)GN * 4));
  int*      NIDX = (int*)     (ws + take((size_t)GN * 4));
  int*      PERM = (int*)     (ws + take((size_t)PK1 * 4));
  float*    SB0  = (float*)   (ws + take((size_t)(PK1 + 16) * 20 * 4));
  float*    SB1  = (float*)   (ws + take((size_t)(PK1 + 16) * 20 * 4));
  float*    SB2  = (float*)   (ws + take((size_t)(PK1 + 16) * 20 * 4));
  float*    WT1  = (float*)   (ws + take((size_t)32 * 8 * 4));
  float*    WT2  = (float*)   (ws + take((size_t)16 * 20 * 4));
  float*    WT3  = (float*)   (ws + take((size_t)16 * 16 * 4));
  float*    WT4  = (float*)   (ws + take((size_t)16 * 12 * 4));
  unsigned* ST   = (unsigned*)(ws + take(256));

  const dim3 B256(256), B1(1);

  auto pool = [&](int n, int f, int in_stride, int k, const float* X, const float* p,
                  float* Xk, int out_stride) {
    score_kernel<<<g(n), B256, 0, stream>>>(X, in_stride, p, f, n, SCORE);
    minmax_init<<<1, B1, 0, stream>>>(ST);
    minmax_reduce<<<g(n), B256, 0, stream>>>(SCORE, n, ST);
    bisect_init<<<1, B1, 0, stream>>>(ST);
    for (int it = 0; it < 25; ++it) {
      bisect_pre<<<1, B1, 0, stream>>>(ST);
      bisect_count<<<g(n), B256, 0, stream>>>(SCORE, n, ST);
      bisect_post<<<1, B1, 0, stream>>>(ST, k);
    }
    zero_u32<<<g(k), B256, 0, stream>>>((unsigned*)PERM, k);
    select_gt<<<g(n), B256, 0, stream>>>(SCORE, n, ST, PERM, k);
    select_rest<<<g(n), B256, 0, stream>>>(SCORE, n, ST, PERM, k);
    zero_u32<<<g(n), B256, 0, stream>>>((unsigned*)KEEP, n);
    zero_u32<<<g(n), B256, 0, stream>>>((unsigned*)NIDX, n);
    int kpad = (k + 15) & ~15;
    zero_f32<<<g((long long)kpad * out_stride), B256, 0, stream>>>(Xk, kpad * out_stride);
    pool_apply<<<g(k), B256, 0, stream>>>(X, in_stride, SCORE, PERM, Xk, out_stride,
                                          KEEP, NIDX, k, f);
    reindex_edges<<<g(GE), B256, 0, stream>>>(ESRC, EDST, EEW, KEEP, NIDX, GE);
  };

  // ---- setup: edge working copies, packed operands ----
  init_edges<<<g(GE), B256, 0, stream>>>(ei, ESRC, EDST, EEW, GE);
  pack_x<<<g((long long)GN * 8), B256, 0, stream>>>(x, XP, GN, 5, 8);
  pack_wt<<<g(32 * 8),  B256, 0, stream>>>(W1, WT1, 5, 20, 8, 32);
  pack_wt<<<g(16 * 20), B256, 0, stream>>>(W2, WT2, 20, 15, 20, 16);
  pack_wt<<<g(16 * 16), B256, 0, stream>>>(W3, WT3, 15, 12, 16, 16);
  pack_wt<<<g(16 * 12), B256, 0, stream>>>(W4, WT4, 12, 10, 12, 16);

  // ---- layer 1: conv(5->20)+relu (hW stride 32), pool to K1 (next lda=20) ----
  run_conv<8, 20, 32>(XP, WT1, b1, GN, 1, HA, HB, ESRC, EDST, EEW, DEG, stream);
  pool(GN, 20, 32, PK1, HA, p1, SB0, 20);

  // ---- layer 2: conv(20->15) (stride 16), pool to K2 (next lda=16) ----
  run_conv<20, 15, 16>(SB0, WT2, b2, PK1, 0, SB1, SB2, ESRC, EDST, EEW, DEG, stream);
  pool(PK1, 15, 16, PK2, SB1, p2, SB0, 16);

  // ---- layer 3: conv(15->12, lda 16) (stride 16), pool to K3 (next lda=12) ----
  run_conv<16, 12, 16>(SB0, WT3, b3, PK2, 0, SB1, SB2, ESRC, EDST, EEW, DEG, stream);
  pool(PK2, 12, 16, PK3, SB1, p3, SB0, 12);

  // ---- layer 4: conv(12->10, lda 12) (stride 16) ----
  run_conv<12, 10, 16>(SB0, WT4, b4, PK3, 0, SB1, SB2, ESRC, EDST, EEW, DEG, stream);

  // ---- head: relu(H @ Wl + bl) -> log_softmax ----
  final_head<<<g(PK3), B256, 0, stream>>>(SB1, 16, Wl, bl, out, PK3, 10);
}